// ShortConv1D_78855599554935
// MI455X (gfx1250) — compile-verified
//
#include <hip/hip_runtime.h>

typedef __attribute__((ext_vector_type(4))) float v4f;

namespace {
constexpr int kDim    = 2048;
constexpr int kSeq    = 4096;
constexpr int kBatch  = 4;
constexpr int kK      = 4;
constexpr int kDV4    = kDim / 4;   // 512 float4 channel-groups
constexpr int kBlock  = 256;        // 8 wave32s per block
constexpr int kTChunk = 64;         // time rows per block
constexpr int kInner  = 8;          // unrolled inner rows
constexpr int kPFRows = 24;         // prefetch distance in rows
}

__global__ __launch_bounds__(kBlock)
void dwconv_causal_k4(const float* __restrict__ x,
                      const float* __restrict__ w,
                      const float* __restrict__ b,
                      float* __restrict__ y)
{
    const int d4 = blockIdx.x * kBlock + threadIdx.x;   // 0..511
    const int d  = d4 << 2;                             // first of 4 channels
    const int t0 = blockIdx.y * kTChunk;
    const int bb = blockIdx.z;

    // w is [D, K] row-major: one b128 per channel row, then transpose in
    // registers to per-tap channel vectors. Tap k multiplies x[t - (K-1) + k].
    const v4f wr0 = *(const v4f*)(w + (size_t)(d + 0) * kK);
    const v4f wr1 = *(const v4f*)(w + (size_t)(d + 1) * kK);
    const v4f wr2 = *(const v4f*)(w + (size_t)(d + 2) * kK);
    const v4f wr3 = *(const v4f*)(w + (size_t)(d + 3) * kK);
    const v4f wk0 = {wr0.x, wr1.x, wr2.x, wr3.x};   // * x[t-3]
    const v4f wk1 = {wr0.y, wr1.y, wr2.y, wr3.y};   // * x[t-2]
    const v4f wk2 = {wr0.z, wr1.z, wr2.z, wr3.z};   // * x[t-1]
    const v4f wk3 = {wr0.w, wr1.w, wr2.w, wr3.w};   // * x[t]
    const v4f bias = *(const v4f*)(b + d);

    const size_t chanBase = (size_t)bb * kSeq * kDim + (size_t)d;
    const float* px = x + chanBase + (size_t)t0 * kDim;
    float*       py = y + chanBase + (size_t)t0 * kDim;

    // Register sliding window; causal zero-pad only at t==0 (t0 is a
    // multiple of kTChunk, so t0 > 0 implies all 3 history rows exist).
    v4f xm3 = {0.f, 0.f, 0.f, 0.f};
    v4f xm2 = {0.f, 0.f, 0.f, 0.f};
    v4f xm1 = {0.f, 0.f, 0.f, 0.f};
    if (t0 > 0) {
        xm3 = __builtin_nontemporal_load((const v4f*)(px - 3 * kDim));
        xm2 = __builtin_nontemporal_load((const v4f*)(px - 2 * kDim));
        xm1 = __builtin_nontemporal_load((const v4f*)(px - 1 * kDim));
    }

    // One lane per 128B cacheline issues the prefetch (lanes are 16B apart).
    const bool pfLane = ((threadIdx.x & 7) == 0);

    for (int o = 0; o < kTChunk / kInner; ++o) {
        if (pfLane) {
            __builtin_prefetch(px + (size_t)kPFRows * kDim, 0, 0);
        }
#pragma unroll
        for (int i = 0; i < kInner; ++i) {
            const v4f xc = __builtin_nontemporal_load((const v4f*)px);
            v4f acc = bias;
            acc += wk0 * xm3;
            acc += wk1 * xm2;
            acc += wk2 * xm1;
            acc += wk3 * xc;
            __builtin_nontemporal_store(acc, (v4f*)py);
            xm3 = xm2; xm2 = xm1; xm1 = xc;
            px += kDim; py += kDim;
        }
    }
}

extern "C" void kernel_launch(void* const* d_in, const int* /*in_sizes*/, int /*n_in*/,
                              void* d_out, int /*out_size*/, void* /*d_ws*/, size_t /*ws_size*/,
                              hipStream_t stream) {
    const float* x = (const float*)d_in[0];
    const float* w = (const float*)d_in[1];
    const float* b = (const float*)d_in[2];
    float*       y = (float*)d_out;

    dim3 grid(kDV4 / kBlock, kSeq / kTChunk, kBatch);  // (2, 64, 4)
    dim3 block(kBlock);
    dwconv_causal_k4<<<grid, block, 0, stream>>>(x, w, b, y);
}